// MultiBoxLoss_24781961298144
// MI455X (gfx1250) — compile-verified
//
#include <hip/hip_runtime.h>
#include <cstdint>
#include <cstddef>

#define CHUNKS 32            // blocks per batch row in kernel 1
#define K1_THREADS 256       // 8 waves (wave32)
#define K2_THREADS 256
#define MAXC_LDS 96          // LDS slot columns per wave buffer (C=81 fits)

#if defined(__HIP_DEVICE_COMPILE__) && defined(__gfx1250__) && \
    __has_builtin(__builtin_amdgcn_global_load_async_to_lds_b32) && \
    __has_builtin(__builtin_amdgcn_s_wait_asynccnt)
#define USE_ASYNC_LDS 1
typedef __attribute__((address_space(1))) int* gint_ptr;
typedef __attribute__((address_space(3))) int* lint_ptr;
#else
#define USE_ASYNC_LDS 0
#endif

__device__ __forceinline__ float wave_max32(float v) {
  #pragma unroll
  for (int off = 16; off > 0; off >>= 1)
    v = fmaxf(v, __shfl_xor(v, off, 32));
  return v;
}
__device__ __forceinline__ float wave_sum32(float v) {
  #pragma unroll
  for (int off = 16; off > 0; off >>= 1)
    v += __shfl_xor(v, off, 32);
  return v;
}

// Shared per-row tail: CE at label (+ GIoU if positive). Lane 0 only.
__device__ __forceinline__ void row_tail(const float* __restrict__ srow,
                                         const float* __restrict__ loc,
                                         const float* __restrict__ tgt,
                                         float* __restrict__ neg_ce,
                                         size_t row, float m, float s,
                                         float& loc_acc, float& pce_acc, int& np_acc)
{
  const float* trow = tgt + row * 5;
  const int label = (int)trow[4];
  const float cl = srow[label];
  const float ce = -(cl - m - logf(s));
  const bool pos = (label > 0);
  // clamp kills -0.0 / tiny negative rounding so uint ordering == float ordering
  neg_ce[row] = pos ? 0.f : fmaxf(ce, 0.f);
  if (pos) {
    pce_acc += ce;
    np_acc  += 1;
    const float* prow = loc + row * 4;
    const float x1 = prow[0], y1 = prow[1], x2 = prow[2], y2 = prow[3];
    const float x1g = trow[0], y1g = trow[1], x2g = trow[2], y2g = trow[3];
    const float area1 = (x2 - x1) * (y2 - y1);
    const float area2 = (x2g - x1g) * (y2g - y1g);
    const float iw = fminf(x2, x2g) - fmaxf(x1, x1g);
    const float ih = fminf(y2, y2g) - fmaxf(y1, y1g);
    const float inter = fmaxf(iw, 0.f) * fmaxf(ih, 0.f);
    const float uni = area1 + area2 - inter;
    const float iou = inter / (uni + 1e-7f);
    const float cw = fmaxf(x2, x2g) - fminf(x1, x1g);
    const float ch = fmaxf(y2, y2g) - fminf(y1, y1g);
    const float areac = cw * ch;
    const float giou = iou - (areac - uni) / (areac + 1e-7f);
    loc_acc += 1.f - giou;
  }
}

// ---------------------------------------------------------------------------
// Kernel 1 (templated on STRIPS = ceil(C/32), 32*(STRIPS-1) < C <= 32*STRIPS):
// per-row log-softmax CE + GIoU. Confidence rows staged global->LDS via CDNA5
// async DMA (ASYNCcnt), double-buffered one row ahead; prefetch two rows ahead.
// ---------------------------------------------------------------------------
template <int STRIPS>
__global__ __launch_bounds__(K1_THREADS)
void mbloss_row_kernel(const float* __restrict__ conf,
                       const float* __restrict__ loc,
                       const float* __restrict__ tgt,
                       float* __restrict__ neg_ce,
                       float* __restrict__ loc_part,
                       float* __restrict__ pce_part,
                       int*   __restrict__ np_part,
                       int N, int C)
{
  constexpr int WPB = K1_THREADS / 32;   // waves per block
  const int b     = blockIdx.y;
  const int chunk = blockIdx.x;
  const int rows_per_chunk = (N + CHUNKS - 1) / CHUNKS;
  const int n0 = chunk * rows_per_chunk;
  const int n1 = (n0 + rows_per_chunk < N) ? (n0 + rows_per_chunk) : N;
  const int wave = threadIdx.x >> 5;
  const int lane = threadIdx.x & 31;

  __shared__ float lds_conf[WPB][2][MAXC_LDS];
  __shared__ float redf[WPB][2];
  __shared__ int   redi[WPB];

  float loc_acc = 0.f, pce_acc = 0.f;
  int   np_acc = 0;
  int   buf = 0;

#if USE_ASYNC_LDS
  // Pipeline prologue: stage the first row into buffer 0.
  {
    const int n = n0 + wave;
    if (n < n1) {
      const float* crow = conf + ((size_t)b * N + n) * (size_t)C;
      #pragma unroll
      for (int j = 0; j < STRIPS; ++j) {
        const int c = lane + 32 * j;
        if (c < C)
          __builtin_amdgcn_global_load_async_to_lds_b32(
              (gint_ptr)(crow + c), (lint_ptr)(&lds_conf[wave][0][c]), 0, 0);
      }
    }
  }
#endif

  for (int n = n0 + wave; n < n1; n += WPB) {
    const size_t row = (size_t)b * (size_t)N + (size_t)n;
    const float* crow = conf + row * (size_t)C;
    const int nn = n + WPB;

#if USE_ASYNC_LDS
    if (nn < n1) {
      const float* crow2 = conf + (row + WPB) * (size_t)C;
      // Prefetch two rows ahead (gfx1250: global_prefetch_b8) so the async
      // stage one row ahead tends to hit WGP$/L2.
      if (nn + WPB < n1) __builtin_prefetch(crow2 + (size_t)WPB * C, 0, 0);
      #pragma unroll
      for (int j = 0; j < STRIPS; ++j) {
        const int c = lane + 32 * j;
        if (c < C)
          __builtin_amdgcn_global_load_async_to_lds_b32(
              (gint_ptr)(crow2 + c), (lint_ptr)(&lds_conf[wave][buf ^ 1][c]), 0, 0);
      }
      // Async loads retire in order: cnt <= STRIPS means the current row's
      // STRIPS older transfers are complete; only next-row loads remain.
      __builtin_amdgcn_s_wait_asynccnt(STRIPS);
    } else {
      __builtin_amdgcn_s_wait_asynccnt(0);
    }
    const float* srow = &lds_conf[wave][buf][0];
#else
    if (nn < n1) __builtin_prefetch(crow + (size_t)WPB * C, 0, 0);
    const float* srow = crow;
#endif

    const float v0 = (lane < C)                      ? srow[lane]      : -INFINITY;
    const float v1 = (STRIPS >= 2 && lane + 32 < C)  ? srow[lane + 32] : -INFINITY;
    const float v2 = (STRIPS >= 3 && lane + 64 < C)  ? srow[lane + 64] : -INFINITY;
    const float m = wave_max32(fmaxf(v0, fmaxf(v1, v2)));
    const float s = wave_sum32(expf(v0 - m) + expf(v1 - m) + expf(v2 - m));

    if (lane == 0)
      row_tail(srow, loc, tgt, neg_ce, row, m, s, loc_acc, pce_acc, np_acc);

    buf ^= 1;
  }

  if (lane == 0) { redf[wave][0] = loc_acc; redf[wave][1] = pce_acc; redi[wave] = np_acc; }
  __syncthreads();
  if (threadIdx.x == 0) {
    float l = 0.f, p = 0.f; int np = 0;
    #pragma unroll
    for (int w = 0; w < WPB; ++w) { l += redf[w][0]; p += redf[w][1]; np += redi[w]; }
    loc_part[b * CHUNKS + chunk] = l;
    pce_part[b * CHUNKS + chunk] = p;
    np_part [b * CHUNKS + chunk] = np;
  }
}

// Generic fallback for C > 96: direct strided global loads, arbitrary C.
__global__ __launch_bounds__(K1_THREADS)
void mbloss_row_kernel_generic(const float* __restrict__ conf,
                               const float* __restrict__ loc,
                               const float* __restrict__ tgt,
                               float* __restrict__ neg_ce,
                               float* __restrict__ loc_part,
                               float* __restrict__ pce_part,
                               int*   __restrict__ np_part,
                               int N, int C)
{
  constexpr int WPB = K1_THREADS / 32;
  const int b     = blockIdx.y;
  const int chunk = blockIdx.x;
  const int rows_per_chunk = (N + CHUNKS - 1) / CHUNKS;
  const int n0 = chunk * rows_per_chunk;
  const int n1 = (n0 + rows_per_chunk < N) ? (n0 + rows_per_chunk) : N;
  const int wave = threadIdx.x >> 5;
  const int lane = threadIdx.x & 31;

  __shared__ float redf[WPB][2];
  __shared__ int   redi[WPB];

  float loc_acc = 0.f, pce_acc = 0.f;
  int   np_acc = 0;

  for (int n = n0 + wave; n < n1; n += WPB) {
    const size_t row = (size_t)b * (size_t)N + (size_t)n;
    const float* crow = conf + row * (size_t)C;

    float mlane = -INFINITY;
    for (int c = lane; c < C; c += 32) mlane = fmaxf(mlane, crow[c]);
    const float m = wave_max32(mlane);
    float slane = 0.f;
    for (int c = lane; c < C; c += 32) slane += expf(crow[c] - m);
    const float s = wave_sum32(slane);

    if (lane == 0)
      row_tail(crow, loc, tgt, neg_ce, row, m, s, loc_acc, pce_acc, np_acc);
  }

  if (lane == 0) { redf[wave][0] = loc_acc; redf[wave][1] = pce_acc; redi[wave] = np_acc; }
  __syncthreads();
  if (threadIdx.x == 0) {
    float l = 0.f, p = 0.f; int np = 0;
    #pragma unroll
    for (int w = 0; w < WPB; ++w) { l += redf[w][0]; p += redf[w][1]; np += redi[w]; }
    loc_part[b * CHUNKS + chunk] = l;
    pce_part[b * CHUNKS + chunk] = p;
    np_part [b * CHUNKS + chunk] = np;
  }
}

// ---------------------------------------------------------------------------
// Kernel 2: per-batch 8-bit-per-pass radix select of the k-th largest neg_ce,
// then exact top-k sum (tie-exact vs argsort-based reference).
// ---------------------------------------------------------------------------
__global__ __launch_bounds__(K2_THREADS)
void mbloss_select_kernel(const float* __restrict__ neg_ce,
                          const float* __restrict__ loc_part,
                          const float* __restrict__ pce_part,
                          const int*   __restrict__ np_part,
                          float* __restrict__ loc_sum,
                          float* __restrict__ pce_sum,
                          float* __restrict__ neg_sum,
                          int*   __restrict__ np_sum,
                          int N)
{
  const int b = blockIdx.x;
  const int tid = threadIdx.x;

  __shared__ unsigned int hist[256];
  __shared__ float fred[K2_THREADS];
  __shared__ unsigned int s_sel;
  __shared__ int s_rem;
  __shared__ int s_k;

  if (tid == 0) {
    float l = 0.f, p = 0.f; int np = 0;
    for (int c = 0; c < CHUNKS; ++c) {
      l += loc_part[b * CHUNKS + c];
      p += pce_part[b * CHUNKS + c];
      np += np_part[b * CHUNKS + c];
    }
    loc_sum[b] = l; pce_sum[b] = p; np_sum[b] = np;
    int k = 3 * np; if (k > N) k = N;
    s_k = k;
  }
  __syncthreads();
  const int k = s_k;
  if (k == 0) { if (tid == 0) neg_sum[b] = 0.f; return; }

  const float* arr = neg_ce + (size_t)b * (size_t)N;
  unsigned int prefix = 0u, pmask = 0u;
  int remaining = k;

  for (int pass = 0; pass < 4; ++pass) {
    const int shift = 24 - 8 * pass;
    hist[tid] = 0u;                       // K2_THREADS == 256
    __syncthreads();
    for (int i = tid; i < N; i += K2_THREADS) {
      const unsigned int v = __float_as_uint(arr[i]);
      if ((v & pmask) == prefix) atomicAdd(&hist[(v >> shift) & 255u], 1u);
    }
    __syncthreads();
    if (tid == 0) {
      int cum = 0, sel = 0, rem = remaining;
      for (int bin = 255; bin >= 0; --bin) {
        const int c = (int)hist[bin];
        if (cum + c >= remaining) { sel = bin; rem = remaining - cum; break; }
        cum += c;
      }
      s_sel = (unsigned int)sel; s_rem = rem;
    }
    __syncthreads();
    prefix |= (s_sel << shift);
    pmask  |= (255u << shift);
    remaining = s_rem;
    __syncthreads();
  }

  const unsigned int tbits = prefix;      // exact bits of the k-th largest value
  float lsum = 0.f;
  for (int i = tid; i < N; i += K2_THREADS) {
    const float f = arr[i];
    if (__float_as_uint(f) > tbits) lsum += f;
  }
  fred[tid] = lsum;
  __syncthreads();
  for (int s2 = K2_THREADS / 2; s2 > 0; s2 >>= 1) {
    if (tid < s2) fred[tid] += fred[tid + s2];
    __syncthreads();
  }
  if (tid == 0)
    neg_sum[b] = fred[0] + (float)remaining * __uint_as_float(tbits);
}

// ---------------------------------------------------------------------------
// Kernel 3: deterministic serial reduction over B -> 3 scalars.
// ---------------------------------------------------------------------------
__global__ void mbloss_final_kernel(const float* __restrict__ loc_sum,
                                    const float* __restrict__ pce_sum,
                                    const float* __restrict__ neg_sum,
                                    const int*   __restrict__ np_sum,
                                    float* __restrict__ out, int B)
{
  if (threadIdx.x == 0 && blockIdx.x == 0) {
    float tot = 0.f, lm = 0.f, cm = 0.f;
    for (int b = 0; b < B; ++b) {
      const float l = loc_sum[b];
      const float c = pce_sum[b] + neg_sum[b];   // ALPHA == 1
      const int np = np_sum[b];
      const float mask = (np > 0) ? 1.f : 0.f;
      const float denom = fmaxf((float)np, 1e-6f);
      tot += (l + c) * mask / denom;
      lm += l; cm += c;
    }
    const float invB = 1.f / (float)B;
    out[0] = tot * invB;
    out[1] = lm * invB;
    out[2] = cm * invB;
  }
}

extern "C" void kernel_launch(void* const* d_in, const int* in_sizes, int n_in,
                              void* d_out, int out_size, void* d_ws, size_t ws_size,
                              hipStream_t stream) {
  const float* conf = (const float*)d_in[0];   // (B,N,C) f32
  const float* loc  = (const float*)d_in[1];   // (B,N,4) f32
  const float* tgt  = (const float*)d_in[2];   // (B,N,5) f32

  const int B = 64;                                      // per reference setup
  const long long bn = (long long)in_sizes[1] / 4;       // B*N
  const int N = (int)(bn / B);
  const int C = (int)((long long)in_sizes[0] / bn);      // 81

  // Workspace layout (all entries fully written before read; no init needed).
  float* neg_ce   = (float*)d_ws;                        // B*N
  float* loc_part = neg_ce + (size_t)B * N;              // B*CHUNKS
  float* pce_part = loc_part + (size_t)B * CHUNKS;       // B*CHUNKS
  int*   np_part  = (int*)(pce_part + (size_t)B * CHUNKS); // B*CHUNKS
  float* loc_sum  = (float*)(np_part + (size_t)B * CHUNKS);// B
  float* pce_sum  = loc_sum + B;                         // B
  float* neg_sum  = pce_sum + B;                         // B
  int*   np_sum   = (int*)(neg_sum + B);                 // B

  dim3 g1(CHUNKS, B);
  if (C <= 32) {
    mbloss_row_kernel<1><<<g1, dim3(K1_THREADS), 0, stream>>>(
        conf, loc, tgt, neg_ce, loc_part, pce_part, np_part, N, C);
  } else if (C <= 64) {
    mbloss_row_kernel<2><<<g1, dim3(K1_THREADS), 0, stream>>>(
        conf, loc, tgt, neg_ce, loc_part, pce_part, np_part, N, C);
  } else if (C <= 96) {
    mbloss_row_kernel<3><<<g1, dim3(K1_THREADS), 0, stream>>>(
        conf, loc, tgt, neg_ce, loc_part, pce_part, np_part, N, C);
  } else {
    mbloss_row_kernel_generic<<<g1, dim3(K1_THREADS), 0, stream>>>(
        conf, loc, tgt, neg_ce, loc_part, pce_part, np_part, N, C);
  }

  mbloss_select_kernel<<<dim3(B), dim3(K2_THREADS), 0, stream>>>(
      neg_ce, loc_part, pce_part, np_part, loc_sum, pce_sum, neg_sum, np_sum, N);

  mbloss_final_kernel<<<dim3(1), dim3(32), 0, stream>>>(
      loc_sum, pce_sum, neg_sum, np_sum, (float*)d_out, B);
}